// RuleAwareGraphConv_58463094833888
// MI455X (gfx1250) — compile-verified
//
#include <hip/hip_runtime.h>
#include <cstddef>
#include <cstdint>

typedef float v2f __attribute__((ext_vector_type(2)));
typedef float v8f __attribute__((ext_vector_type(8)));

#define DIN 128
#define NACT 4
#define A_STRIDE 132   // 64-row LDS tile, padded: 132 mod 64 banks -> conflict-free column reads

// ---------------- helpers ----------------

__device__ __forceinline__ float wred32(float v) {
#pragma unroll
  for (int o = 16; o > 0; o >>= 1) v += __shfl_xor(v, o, 32);
  return v;
}

// order-preserving float <-> uint mapping (for atomicMax on floats incl. negatives)
__device__ __forceinline__ unsigned fmap(float f) {
  unsigned u = __float_as_uint(f);
  return (u & 0x80000000u) ? ~u : (u | 0x80000000u);
}
__device__ __forceinline__ float funmap(unsigned u) {
  u = (u & 0x80000000u) ? (u & 0x7fffffffu) : ~u;
  return __uint_as_float(u);
}

// ---------------- WMMA inner kernel (f32 16x16x4) ----------------
// Alds: points at this K-chunk's column base of the 64 x 128 A tile (stride A_STRIDE).
// Bch : 32 x 128 chunk of B (k-major).
// Per ISA 7.12.2: A lanes0-15 hold M=0..15 K={k,k+1}; lanes16-31 M=0..15 K={k+2,k+3}.
//                 B lanes0-15 hold N=0..15 K={k,k+1}; lanes16-31 N=0..15 K={k+2,k+3}.
__device__ __forceinline__ void wmma_kchunk(const float* __restrict__ Alds,
                                            const float* __restrict__ Bch,
                                            int rowSel, int kofs, int col16,
                                            v8f acc[8]) {
  const float* arow = Alds + rowSel * A_STRIDE + kofs;
#pragma unroll
  for (int k = 0; k < 32; k += 4) {
    float2 af = *(const float2*)(arow + k);   // 8B aligned: even element index
    v2f a; a.x = af.x; a.y = af.y;
#pragma unroll
    for (int t = 0; t < 8; ++t) {
      v2f b;
      b.x = Bch[(k + kofs) * DIN + t * 16 + col16];
      b.y = Bch[(k + kofs + 1) * DIN + t * 16 + col16];
      acc[t] = __builtin_amdgcn_wmma_f32_16x16x4_f32(false, a, false, b,
                                                     (short)0, acc[t], false, false);
    }
  }
}

// ---------------- dense GEMM: C[M,128] = A[M,128] @ B[128,128] (+bias) ----------------
__global__ void __launch_bounds__(128)
gemm128_kernel(const float* __restrict__ Amat, const float* __restrict__ Bmat,
               const float* __restrict__ bias, float* __restrict__ C, int M) {
  __shared__ float Alds[64 * A_STRIDE];
  __shared__ float Bch[32 * DIN];
  const int tid = threadIdx.x, warp = tid >> 5, lane = tid & 31;
  const int row0 = blockIdx.x * 64;

  for (int i = warp; i < 64; i += 4) {
    int gr = row0 + i;
    float4 v = make_float4(0.f, 0.f, 0.f, 0.f);
    if (gr < M) v = *(const float4*)&Amat[(size_t)gr * DIN + lane * 4];
    *(float4*)&Alds[i * A_STRIDE + lane * 4] = v;
  }

  const int hi = lane >> 4, col16 = lane & 15, kofs = hi * 2;
  const int rowSel = warp * 16 + (lane & 15);
  v8f acc[8] = {};

  for (int kc = 0; kc < 4; ++kc) {
    __syncthreads();
    const float4* Bsrc = (const float4*)(Bmat + (size_t)kc * 32 * DIN);
    for (int j = tid; j < 32 * DIN / 4; j += 128) ((float4*)Bch)[j] = Bsrc[j];
    __syncthreads();
    wmma_kchunk(Alds + kc * 32, Bch, rowSel, kofs, col16, acc);
  }

#pragma unroll
  for (int t = 0; t < 8; ++t) {
    int col = t * 16 + col16;
    float bv = bias ? bias[col] : 0.0f;
#pragma unroll
    for (int i = 0; i < 8; ++i) {
      int gr = row0 + warp * 16 + hi * 8 + i;
      if (gr < M) C[(size_t)gr * DIN + col] = acc[t][i] + bv;
    }
  }
}

// ------------- edge-message GEMM: per-relation tiles, gather + scatter-atomic -------------
__global__ void __launch_bounds__(128)
msg_gemm_kernel(const float* __restrict__ x, const float* __restrict__ W_r,
                const int* __restrict__ perm, const int* __restrict__ src,
                const int* __restrict__ dst, const float* __restrict__ attnm,
                const int* __restrict__ offs, float* __restrict__ outacc, int R) {
  __shared__ float Alds[64 * A_STRIDE];
  __shared__ float Bch[32 * DIN];
  __shared__ float attn_s[64];
  __shared__ int dst_s[64];

  const int tid = threadIdx.x, warp = tid >> 5, lane = tid & 31;

  // flat tile id -> (relation, local tile)
  int bid = blockIdx.x;
  int rel = -1, estart = 0, ecnt = 0, accum = 0;
  for (int rr = 0; rr < R; ++rr) {
    int s0 = offs[rr], s1 = offs[rr + 1];
    int tiles = (s1 - s0 + 63) >> 6;
    if (bid < accum + tiles) {
      rel = rr;
      estart = s0 + (bid - accum) * 64;
      ecnt = s1 - estart; if (ecnt > 64) ecnt = 64;
      break;
    }
    accum += tiles;
  }
  if (rel < 0) return;

  // gather A = x[src[perm[estart..]]] (zero pad), plus attn/dst side data
  for (int i = warp; i < 64; i += 4) {
    float4 v = make_float4(0.f, 0.f, 0.f, 0.f);
    if (i < ecnt) {
      int e = perm[estart + i];
      v = *(const float4*)&x[(size_t)src[e] * DIN + lane * 4];
      if (lane == 0) { attn_s[i] = attnm[e]; dst_s[i] = dst[e]; }
    } else if (lane == 0) { attn_s[i] = 0.0f; dst_s[i] = 0; }
    *(float4*)&Alds[i * A_STRIDE + lane * 4] = v;
  }

  const int hi = lane >> 4, col16 = lane & 15, kofs = hi * 2;
  const int rowSel = warp * 16 + (lane & 15);
  v8f acc[8] = {};
  const float* Bmat = W_r + (size_t)rel * DIN * DIN;  // W_r[rel][k][n], k-major == B layout

  for (int kc = 0; kc < 4; ++kc) {
    __syncthreads();
    const float4* Bsrc = (const float4*)(Bmat + (size_t)kc * 32 * DIN);
    for (int j = tid; j < 32 * DIN / 4; j += 128) ((float4*)Bch)[j] = Bsrc[j];
    __syncthreads();
    wmma_kchunk(Alds + kc * 32, Bch, rowSel, kofs, col16, acc);
  }

#pragma unroll
  for (int t = 0; t < 8; ++t) {
    int col = t * 16 + col16;
#pragma unroll
    for (int i = 0; i < 8; ++i) {
      int lrow = warp * 16 + hi * 8 + i;
      if (lrow < ecnt) {
        float v = acc[t][i] * attn_s[lrow];
        __hip_atomic_fetch_add(&outacc[(size_t)dst_s[lrow] * DIN + col], v,
                               __ATOMIC_RELAXED, __HIP_MEMORY_SCOPE_AGENT);
      }
    }
  }
}

// ---------------- small / elementwise kernels ----------------

__global__ void fill_u32_kernel(unsigned* p, unsigned v, size_t n) {
  size_t i = (size_t)blockIdx.x * 256 + threadIdx.x;
  if (i < n) p[i] = v;
}

__global__ void initout_kernel(float* o, const float* __restrict__ bias, size_t n) {
  size_t i = (size_t)blockIdx.x * 256 + threadIdx.x;
  if (i < n) o[i] = bias[i & (DIN - 1)];
}

// h_rel[r][d] = mean_o W_r[r][d][o]
__global__ void hrel_kernel(const float* __restrict__ W_r, float* __restrict__ hrel, int RD) {
  int t = blockIdx.x * 256 + threadIdx.x;
  if (t >= RD) return;
  const float4* p = (const float4*)(W_r + (size_t)t * DIN);
  float s = 0.f;
  for (int o = 0; o < DIN / 4; ++o) { float4 v = p[o]; s += v.x + v.y + v.z + v.w; }
  hrel[t] = s * (1.0f / DIN);
}

// Vrule[a][o] = rule_emb[rule_ids[a]] . Wk3[:,o]
__global__ void vrule_kernel(const float* __restrict__ rule_emb, const int* __restrict__ rule_ids,
                             const float* __restrict__ Wk3, float* __restrict__ Vrule) {
  int t = blockIdx.x * 256 + threadIdx.x;
  if (t >= NACT * DIN) return;
  int a = t >> 7, o = t & 127;
  const float* he = rule_emb + (size_t)rule_ids[a] * DIN;
  float s = 0.f;
  for (int d = 0; d < DIN; ++d) s += he[d] * Wk3[(size_t)d * DIN + o];
  Vrule[t] = s;
}

// one wave per edge: score_base + per-rule scores, write scores, atomicMax per segment
__global__ void __launch_bounds__(256)
score_kernel(const float* __restrict__ Q, const float* __restrict__ Kn,
             const float* __restrict__ Krel, const float* __restrict__ Vrule,
             const int* __restrict__ src, const int* __restrict__ dst,
             const int* __restrict__ et, float* __restrict__ scores,
             unsigned* __restrict__ mseg, int E, int N) {
  int e = blockIdx.x * 8 + (threadIdx.x >> 5);
  if (e >= E) return;
  int lane = threadIdx.x & 31;
  int d = dst[e], s = src[e], r = et[e];
  const float4 q  = *(const float4*)&Q[(size_t)d * DIN + lane * 4];
  const float4 kn = *(const float4*)&Kn[(size_t)s * DIN + lane * 4];
  const float4 kr = *(const float4*)&Krel[(size_t)r * DIN + lane * 4];
  float pb = q.x * (kn.x + kr.x) + q.y * (kn.y + kr.y) +
             q.z * (kn.z + kr.z) + q.w * (kn.w + kr.w);
  float pa[NACT];
#pragma unroll
  for (int a = 0; a < NACT; ++a) {
    const float4 v = *(const float4*)&Vrule[a * DIN + lane * 4];
    pa[a] = q.x * v.x + q.y * v.y + q.z * v.z + q.w * v.w;
  }
  pb = wred32(pb);
#pragma unroll
  for (int a = 0; a < NACT; ++a) pa[a] = wred32(pa[a]);
  if (lane == 0) {
    const float scale = 0.088388347648318447f;  // 1/sqrt(128)
    size_t seg = ((size_t)r * N + d) * NACT;
#pragma unroll
    for (int a = 0; a < NACT; ++a) {
      float sc = (pb + pa[a]) * scale;
      scores[(size_t)e * NACT + a] = sc;
      atomicMax(&mseg[seg + a], fmap(sc));
    }
  }
}

__global__ void exp_kernel(const int* __restrict__ dst, const int* __restrict__ et,
                           float* __restrict__ scores, const unsigned* __restrict__ mseg,
                           float* __restrict__ ssum, int E, int N) {
  size_t t = (size_t)blockIdx.x * 256 + threadIdx.x;
  if (t >= (size_t)E * NACT) return;
  int e = (int)(t >> 2), a = (int)(t & 3);
  size_t seg = ((size_t)et[e] * N + dst[e]) * NACT + a;
  float ex = __expf(scores[t] - funmap(mseg[seg]));
  scores[t] = ex;
  __hip_atomic_fetch_add(&ssum[seg], ex, __ATOMIC_RELAXED, __HIP_MEMORY_SCOPE_AGENT);
}

__global__ void attn_kernel(const int* __restrict__ dst, const int* __restrict__ et,
                            const float* __restrict__ scores, const float* __restrict__ ssum,
                            float* __restrict__ attnm, int E, int N) {
  int e = blockIdx.x * 256 + threadIdx.x;
  if (e >= E) return;
  size_t seg = ((size_t)et[e] * N + dst[e]) * NACT;
  float s = 0.f;
#pragma unroll
  for (int a = 0; a < NACT; ++a) s += scores[(size_t)e * NACT + a] / ssum[seg + a];
  attnm[e] = s * (1.0f / NACT);
}

__global__ void hist_kernel(const int* __restrict__ et, int* cnt, int E) {
  int e = blockIdx.x * 256 + threadIdx.x;
  if (e < E) atomicAdd(&cnt[et[e]], 1);
}

__global__ void scan_kernel(const int* __restrict__ cnt, int* offs, int* cursor, int R) {
  if (threadIdx.x == 0) {
    int a = 0;
    for (int r = 0; r < R; ++r) { offs[r] = a; cursor[r] = a; a += cnt[r]; }
    offs[R] = a;
  }
}

__global__ void scatter_kernel(const int* __restrict__ et, int* cursor, int* perm, int E) {
  int e = blockIdx.x * 256 + threadIdx.x;
  if (e < E) {
    int p = atomicAdd(&cursor[et[e]], 1);
    perm[p] = e;
  }
}

// one wave per node: LayerNorm + ReLU
__global__ void __launch_bounds__(256)
ln_kernel(const float* __restrict__ outacc, const float* __restrict__ gamma,
          const float* __restrict__ beta, float* __restrict__ out, int N) {
  int n = blockIdx.x * 8 + (threadIdx.x >> 5);
  if (n >= N) return;
  int lane = threadIdx.x & 31;
  float4 v = *(const float4*)&outacc[(size_t)n * DIN + lane * 4];
  float s  = v.x + v.y + v.z + v.w;
  float sq = v.x * v.x + v.y * v.y + v.z * v.z + v.w * v.w;
  s = wred32(s); sq = wred32(sq);
  float mean = s * (1.0f / DIN);
  float var  = sq * (1.0f / DIN) - mean * mean;
  float inv  = rsqrtf(var + 1e-5f);
  float4 g = *(const float4*)&gamma[lane * 4];
  float4 b = *(const float4*)&beta[lane * 4];
  float4 o;
  o.x = fmaxf((v.x - mean) * inv * g.x + b.x, 0.0f);
  o.y = fmaxf((v.y - mean) * inv * g.y + b.y, 0.0f);
  o.z = fmaxf((v.z - mean) * inv * g.z + b.z, 0.0f);
  o.w = fmaxf((v.w - mean) * inv * g.w + b.w, 0.0f);
  *(float4*)&out[(size_t)n * DIN + lane * 4] = o;
}

// ---------------- host launcher ----------------

extern "C" void kernel_launch(void* const* d_in, const int* in_sizes, int n_in,
                              void* d_out, int out_size, void* d_ws, size_t ws_size,
                              hipStream_t stream) {
  const float* x        = (const float*)d_in[0];
  const float* W_r      = (const float*)d_in[1];
  const float* Wq_w     = (const float*)d_in[2];
  const float* Wq_b     = (const float*)d_in[3];
  const float* Wk_w     = (const float*)d_in[4];
  const float* Wk_b     = (const float*)d_in[5];
  const float* rule_emb = (const float*)d_in[6];
  const float* bias     = (const float*)d_in[7];
  const float* gamma    = (const float*)d_in[8];
  const float* beta     = (const float*)d_in[9];
  const int* src      = (const int*)d_in[10];
  const int* dst      = (const int*)d_in[11];
  const int* et       = (const int*)d_in[12];
  const int* rule_ids = (const int*)d_in[13];

  const int N = in_sizes[0] / DIN;
  const int R = in_sizes[1] / (DIN * DIN);
  const int E = in_sizes[10];

  const float* Wk1 = Wk_w;                    // [128,128]
  const float* Wk2 = Wk_w + DIN * DIN;        // [128,128]
  const float* Wk3 = Wk_w + 2 * DIN * DIN;    // [128,128]

  // workspace carve-out (256B aligned)
  size_t off = 0;
  char* base = (char*)d_ws;
  auto alloc = [&](size_t bytes) -> void* {
    void* p = base + off;
    off += (bytes + 255) & ~(size_t)255;
    return p;
  };
  float*    Q      = (float*)alloc((size_t)N * DIN * 4);
  float*    Kn     = (float*)alloc((size_t)N * DIN * 4);
  float*    Krel   = (float*)alloc((size_t)R * DIN * 4);
  float*    Vrule  = (float*)alloc((size_t)NACT * DIN * 4);
  float*    hrel   = (float*)alloc((size_t)R * DIN * 4);
  float*    scores = (float*)alloc((size_t)E * NACT * 4);
  unsigned* mseg   = (unsigned*)alloc((size_t)R * N * NACT * 4);
  float*    ssum   = (float*)alloc((size_t)R * N * NACT * 4);
  float*    attnm  = (float*)alloc((size_t)E * 4);
  float*    outacc = (float*)alloc((size_t)N * DIN * 4);
  int*      perm   = (int*)alloc((size_t)E * 4);
  int*      cnt    = (int*)alloc(64 * 4);
  int*      offs   = (int*)alloc(65 * 4);
  int*      cursor = (int*)alloc(64 * 4);
  (void)ws_size; (void)n_in; (void)out_size;

  const size_t nsegA = (size_t)R * N * NACT;
  auto blks = [](size_t n, int b) { return (unsigned)((n + b - 1) / b); };

  // init scratch (every call: deterministic across graph replays)
  fill_u32_kernel<<<blks(nsegA, 256), 256, 0, stream>>>(mseg, 0u, nsegA);
  fill_u32_kernel<<<blks(nsegA, 256), 256, 0, stream>>>((unsigned*)ssum, 0u, nsegA);
  fill_u32_kernel<<<1, 256, 0, stream>>>((unsigned*)cnt, 0u, 64);
  initout_kernel<<<blks((size_t)N * DIN, 256), 256, 0, stream>>>(outacc, bias, (size_t)N * DIN);

  // small precomputes
  hrel_kernel<<<blks((size_t)R * DIN, 256), 256, 0, stream>>>(W_r, hrel, R * DIN);
  gemm128_kernel<<<blks(R, 64), 128, 0, stream>>>(hrel, Wk2, nullptr, Krel, R);
  vrule_kernel<<<blks(NACT * DIN, 256), 256, 0, stream>>>(rule_emb, rule_ids, Wk3, Vrule);

  // Q = x@Wq + Wq_b ; Kn = x@Wk1 + Wk_b  (bias folded)
  gemm128_kernel<<<blks(N, 64), 128, 0, stream>>>(x, Wq_w, Wq_b, Q, N);
  gemm128_kernel<<<blks(N, 64), 128, 0, stream>>>(x, Wk1, Wk_b, Kn, N);

  // scores + segment max
  score_kernel<<<blks(E, 8), 256, 0, stream>>>(Q, Kn, Krel, Vrule, src, dst, et,
                                               scores, mseg, E, N);
  // exp + segment sum
  exp_kernel<<<blks((size_t)E * NACT, 256), 256, 0, stream>>>(dst, et, scores, mseg, ssum, E, N);
  // attn mean
  attn_kernel<<<blks(E, 256), 256, 0, stream>>>(dst, et, scores, ssum, attnm, E, N);

  // bucket edges by relation
  hist_kernel<<<blks(E, 256), 256, 0, stream>>>(et, cnt, E);
  scan_kernel<<<1, 32, 0, stream>>>(cnt, offs, cursor, R);
  scatter_kernel<<<blks(E, 256), 256, 0, stream>>>(et, cursor, perm, E);

  // heavy stage: per-relation WMMA GEMM, scaled by attn, scatter-add into outacc
  unsigned msgTiles = blks(E, 64) + R;  // upper bound on sum of per-relation tile counts
  msg_gemm_kernel<<<msgTiles, 128, 0, stream>>>(x, W_r, perm, src, dst, attnm,
                                                offs, outacc, R);

  // LayerNorm + ReLU -> d_out
  ln_kernel<<<blks(N, 8), 256, 0, stream>>>(outacc, gamma, beta, (float*)d_out, N);
}